// Gen2SingleInput_44238163148998
// MI455X (gfx1250) — compile-verified
//
#include <hip/hip_runtime.h>

// Problem constants (match reference)
#define T_STEPS 1024
#define BATCH   8
#define INDIM   512
#define DV      64
#define DKEY    64
#define ROWS    (T_STEPS * BATCH)   // 8192 rows, row = t*B + b
#define NCOLS   192                 // [Wv | Wk | Wa] stacked
#define EPSF    1e-8f
#define CHUNK   64
#define NCHUNK  (T_STEPS / CHUNK)   // 16

typedef __attribute__((ext_vector_type(2))) float v2f;
typedef __attribute__((ext_vector_type(8))) float v8f;
typedef __attribute__((ext_vector_type(4))) unsigned int u32x4;
typedef __attribute__((ext_vector_type(8))) int i32x8;
typedef __attribute__((ext_vector_type(4))) int i32x4;

// ---------------------------------------------------------------------------
// Kernel 1: fused projections via V_WMMA_F32_16X16X4_F32  (unchanged: emits
// 4x v_wmma_f32_16x16x4_f32 per unrolled loop body).
// ---------------------------------------------------------------------------
__global__ __launch_bounds__(256) void proj_wmma_kernel(
    const float* __restrict__ X,
    const float* __restrict__ Wv, const float* __restrict__ bv,
    const float* __restrict__ Wk, const float* __restrict__ bk,
    const float* __restrict__ Wa, const float* __restrict__ ba,
    float* __restrict__ V, float* __restrict__ Kp, float* __restrict__ PA)
{
    const int wave = (blockIdx.x * blockDim.x + threadIdx.x) >> 5;
    const int lane = threadIdx.x & 31;
    const int tilesN = NCOLS / 16;              // 12 column tiles
    const int tm = wave / tilesN;               // 0..511 row tile
    const int tn = wave % tilesN;               // 0..11 column tile
    const int row0 = tm * 16;
    const int col0 = tn * 16;
    const int hb = lane >> 4;                   // selects K-pair (0/1)
    const int l  = lane & 15;

    const int col   = col0 + l;                 // 0..191
    const int which = col >> 6;                 // 0=V, 1=K, 2=alpha-proj
    const int c64   = col & 63;
    const float* W    = (which == 0) ? Wv : ((which == 1) ? Wk : Wa);
    const float* bias = (which == 0) ? bv : ((which == 1) ? bk : ba);
    float*       dst  = (which == 0) ? V  : ((which == 1) ? Kp : PA);

    const float* arow = X + (size_t)(row0 + l) * INDIM + 2 * hb;  // A[M=l][K=2hb..]
    const float* brow = W + (size_t)c64 * INDIM + 2 * hb;         // B[K][N=l]

    v8f acc = {};
    #pragma unroll 4
    for (int k = 0; k < INDIM; k += 4) {
        v2f a, b;
        a.x = arow[k];  a.y = arow[k + 1];
        b.x = brow[k];  b.y = brow[k + 1];
        acc = __builtin_amdgcn_wmma_f32_16x16x4_f32(
            false, a, false, b, (short)0, acc, false, false);
    }

    const float bb = bias[c64];
    #pragma unroll
    for (int r = 0; r < 8; ++r) {
        const int row = row0 + r + 8 * hb;      // C layout: lanes 16-31 hold M=r+8
        dst[(size_t)row * 64 + c64] = acc[r] + bb;
    }
}

// ---------------------------------------------------------------------------
// Kernel 2: per-(b,n) serial decay scan over T (unchanged).
// ---------------------------------------------------------------------------
__global__ __launch_bounds__(256) void decay_kernel(
    const float* __restrict__ PA, const float* __restrict__ Kp,
    float* __restrict__ KD, float* __restrict__ LC)
{
    const int tid = blockIdx.x * blockDim.x + threadIdx.x;   // b*64 + n
    if (tid >= BATCH * DKEY) return;
    float run = 0.f;
    for (int t = 0; t < T_STEPS; ++t) {
        const size_t idx = (size_t)t * (BATCH * DKEY) + tid;
        const float z     = PA[idx];
        const float alpha = 1.f / (1.f + __expf(-z));
        run += __logf(fmaxf(alpha, EPSF));
        LC[idx] = run;
    }
    const float inv = 1.f / (__expf(run) + EPSF);
    for (int t = 0; t < T_STEPS; ++t) {
        const size_t idx = (size_t)t * (BATCH * DKEY) + tid;
        KD[idx] = Kp[idx] * (__expf(LC[idx]) * inv);
    }
}

// ---------------------------------------------------------------------------
// TDM: issue a 2-D tile load (tile_h rows x tile_w f32, row stride
// row_stride_elems) from global memory into LDS at byte offset lds_off.
// Descriptor per CDNA5 ISA 8.3/8.4: group0 = control/lds/global addr,
// group1 = dims/tile/strides, groups 2-3 zero (2-D tensor).
// ---------------------------------------------------------------------------
__device__ __forceinline__ void tdm_load_2d(
    unsigned lds_off, const float* gptr,
    unsigned tile_w, unsigned tile_h, unsigned row_stride_elems)
{
    const unsigned long long ga = (unsigned long long)(uintptr_t)gptr;

    u32x4 g0;
    g0[0] = 1u;                                   // count=1 (valid user D#)
    g0[1] = lds_off;                              // lds_addr (bytes)
    g0[2] = (unsigned)(ga & 0xffffffffu);         // global_addr[31:0]
    g0[3] = (unsigned)((ga >> 32) & 0x01ffffffu)  // global_addr[56:32]
          | (2u << 30);                           // type = 2 ("image")

    i32x8 g1;
    g1[0] = (int)(2u << 16);                      // wg_mask=0, data_size=2 (4B)
    g1[1] = (int)(tile_w << 16);                  // tensor_dim0[15:0] @ bits 63:48
    g1[2] = (int)(tile_h << 16);                  // tensor_dim1[15:0] @ bits 95:80
    g1[3] = (int)(tile_w << 16);                  // tile_dim0 @ bits 127:112
    g1[4] = (int)tile_h;                          // tile_dim1 @ bits 143:128
    g1[5] = (int)row_stride_elems;                // tensor_dim0_stride[31:0]
    g1[6] = 0;                                    // stride hi / dim1_stride lo
    g1[7] = 0;

    i32x4 g2 = {0, 0, 0, 0};                      // 2-D: higher dims unused
    i32x4 g3 = {0, 0, 0, 0};

#if defined(__clang_major__) && (__clang_major__ >= 23)
    i32x8 gz = {};
    __builtin_amdgcn_tensor_load_to_lds(g0, g1, g2, g3, gz, 0);
#else
    __builtin_amdgcn_tensor_load_to_lds(g0, g1, g2, g3, 0);
#endif
}

// Fill sV (64t x 16d) and sKD (64t x 64n) tiles via TDM from wave 0,
// then rendezvous the whole workgroup.
__device__ __forceinline__ void load_tiles_tdm(
    const float* __restrict__ V, const float* __restrict__ KD,
    int t0, int b, int dt)
{
    if (threadIdx.x < 32) {                       // one wave issues both DMAs
        const float* vsrc  = V  + ((size_t)t0 * BATCH + b) * 64 + dt * 16;
        const float* kdsrc = KD + ((size_t)t0 * BATCH + b) * 64;
        tdm_load_2d(0u,    vsrc,  16u, (unsigned)CHUNK, (unsigned)(BATCH * 64));
        tdm_load_2d(4096u, kdsrc, 64u, (unsigned)CHUNK, (unsigned)(BATCH * 64));
        __builtin_amdgcn_s_wait_tensorcnt(0);     // TENSORcnt == 0
    }
    __syncthreads();
}

// ---------------------------------------------------------------------------
// Kernels 3-5: chunked parallel scan of weighted(t) = v(t,d)*kd(t,n).
// Block = one (chunk, b, d-tile); 1024 threads = 16 d x 64 n.
// Dynamic LDS: [0,4096) = sV, [4096,20480) = sKD (matches TDM lds_addr).
// ---------------------------------------------------------------------------
__global__ __launch_bounds__(1024) void chunk_tot_kernel(
    const float* __restrict__ V, const float* __restrict__ KD,
    float* __restrict__ TOT)
{
    extern __shared__ float smem[];
    float (*sV)[16]   = (float (*)[16])smem;               // lds byte off 0
    float (*sKD)[DKEY] = (float (*)[DKEY])(smem + CHUNK * 16); // byte off 4096

    const int c = blockIdx.x, b = blockIdx.y, dt = blockIdx.z;
    const int tid = threadIdx.x;
    const int dl = tid >> 6;                    // 0..15 (broadcast across wave)
    const int n  = tid & 63;                    // conflict-free LDS reads
    load_tiles_tdm(V, KD, c * CHUNK, b, dt);

    float s = 0.f;
    #pragma unroll 8
    for (int t = 0; t < CHUNK; ++t) s += sV[t][dl] * sKD[t][n];
    const int d = dt * 16 + dl;
    TOT[(((size_t)c * BATCH + b) * 64 + d) * 64 + n] = s;
}

__global__ __launch_bounds__(1024) void chunk_scan_kernel(float* __restrict__ TOT)
{
    const int tid = blockIdx.x * blockDim.x + threadIdx.x;  // b*4096 + d*64 + n
    const int b = tid >> 12, dn = tid & 4095;
    float s = 0.f;
    #pragma unroll
    for (int c = 0; c < NCHUNK; ++c) {
        const size_t idx = (((size_t)c * BATCH + b) << 12) + dn;
        const float v = TOT[idx];
        TOT[idx] = s;                           // exclusive prefix
        s += v;
    }
}

__global__ __launch_bounds__(1024) void scan_out_kernel(
    const float* __restrict__ V, const float* __restrict__ KD,
    const float* __restrict__ TOT, float* __restrict__ S)
{
    extern __shared__ float smem[];
    float (*sV)[16]   = (float (*)[16])smem;
    float (*sKD)[DKEY] = (float (*)[DKEY])(smem + CHUNK * 16);

    const int c = blockIdx.x, b = blockIdx.y, dt = blockIdx.z;
    const int tid = threadIdx.x;
    const int dl = tid >> 6;
    const int n  = tid & 63;
    const int t0 = c * CHUNK;
    load_tiles_tdm(V, KD, t0, b, dt);

    const int d = dt * 16 + dl;
    float s = TOT[(((size_t)c * BATCH + b) * 64 + d) * 64 + n];
    #pragma unroll 4
    for (int t = 0; t < CHUNK; ++t) {
        s += sV[t][dl] * sKD[t][n];
        // n is fastest axis -> 128B contiguous per wave32 store
        S[(((size_t)(t0 + t) * BATCH + b) * 64 + d) * 64 + n] = s;
    }
}

// ---------------------------------------------------------------------------
extern "C" void kernel_launch(void* const* d_in, const int* in_sizes, int n_in,
                              void* d_out, int out_size, void* d_ws, size_t ws_size,
                              hipStream_t stream)
{
    (void)in_sizes; (void)n_in; (void)out_size; (void)ws_size;
    const float* x  = (const float*)d_in[0];
    const float* Wv = (const float*)d_in[1];
    const float* bv = (const float*)d_in[2];
    const float* Wk = (const float*)d_in[3];
    const float* bk = (const float*)d_in[4];
    const float* Wa = (const float*)d_in[5];
    const float* ba = (const float*)d_in[6];
    float* out = (float*)d_out;

    // Workspace carve-up (10 MiB total): V | K | PA | KD | shared(LC/TOT)
    const size_t N64 = (size_t)ROWS * 64;       // 524288 floats = 2 MiB
    float* V  = (float*)d_ws;
    float* Kp = V  + N64;
    float* PA = Kp + N64;
    float* KD = PA + N64;
    float* SH = KD + N64;   // LC (kernel 2) then TOT (kernels 3-5), time-shared

    proj_wmma_kernel<<<768, 256, 0, stream>>>(x, Wv, bv, Wk, bk, Wa, ba, V, Kp, PA);
    decay_kernel<<<2, 256, 0, stream>>>(PA, Kp, KD, SH);

    const size_t smem = (size_t)(CHUNK * 16 + CHUNK * DKEY) * sizeof(float); // 20 KB
    dim3 g(NCHUNK, BATCH, DV / 16);             // 16 x 8 x 4 = 512 blocks
    chunk_tot_kernel<<<g, 1024, smem, stream>>>(V, KD, SH);
    chunk_scan_kernel<<<32, 1024, 0, stream>>>(SH);
    scan_out_kernel<<<g, 1024, smem, stream>>>(V, KD, SH, out);
}